// ITLRegularizer_39865886441997
// MI455X (gfx1250) — compile-verified
//
#include <hip/hip_runtime.h>
#include <math.h>

typedef __attribute__((ext_vector_type(2))) float v2f;
typedef __attribute__((ext_vector_type(8))) float v8f;

namespace {
constexpr int kN            = 8192;
constexpr int kD            = 3;
constexpr int kTiles        = kN / 16;                 // 512 16-wide tiles per dim
constexpr int kChunk        = 64;                      // column tiles per wave-job
constexpr int kChunksPerRow = kTiles / kChunk;         // 8
constexpr int kJobsPerMat   = kTiles * kChunksPerRow;  // 4096
constexpr int kTotalJobs    = 3 * kJobsPerMat;         // 12288
constexpr int kBlock        = 256;                     // 8 waves (wave32)
constexpr int kWavesPerBlk  = kBlock / 32;
constexpr int kGrid         = kTotalJobs / kWavesPerBlk;  // 1536
}  // namespace

// One wave per (matrix, row-tile, column-chunk) job. For each 16x16 tile the
// wave does one V_WMMA_F32_16X16X4_F32 (f32 dot products, K padded 3->4),
// then 8 exps per lane on the D accumulator, summing G = exp(-d2/ks).
__global__ __launch_bounds__(kBlock)
void itl_gram_accum(const float* __restrict__ X,
                    const float* __restrict__ Z,
                    const float* __restrict__ ks_p,
                    float* __restrict__ partials) {
  const int lane = threadIdx.x & 31;
  const int wave = blockIdx.x * kWavesPerBlk + (threadIdx.x >> 5);
  if (wave >= kTotalJobs) return;  // wave-uniform; grid is exact

  const int m       = wave / kJobsPerMat;  // 0: XX, 1: ZZ, 2: XZ
  const int rem     = wave - m * kJobsPerMat;
  const int rowTile = rem >> 3;                  // / kChunksPerRow
  const int chunk   = rem & (kChunksPerRow - 1);

  const float* __restrict__ A = (m == 1) ? Z : X;
  const float* __restrict__ B = (m == 0) ? X : Z;

  // exp(-d2/ks) = exp2(d2 * (-1/(ks*ln2)))
  const float expScale = -1.442695040888963f / ks_p[0];

  const bool hi  = lane >= 16;
  const int  l15 = lane & 15;

  // A-matrix 16x4 f32 VGPR layout: lanes 0-15 {K0,K1}, lanes 16-31 {K2,pad}.
  v2f a;
  {
    const int r  = rowTile * 16 + l15;
    const float a0 = A[r * 3 + 0];
    const float a1 = A[r * 3 + 1];
    const float a2 = A[r * 3 + 2];
    a.x = hi ? a2 : a0;
    a.y = hi ? 0.0f : a1;
  }

  // Squared norms of the 8 rows this lane owns in C/D (row = rb + v).
  float rn[8];
  {
    const int rb = rowTile * 16 + (hi ? 8 : 0);
#pragma unroll
    for (int v = 0; v < 8; ++v) {
      const float r0 = A[(rb + v) * 3 + 0];
      const float r1 = A[(rb + v) * 3 + 1];
      const float r2 = A[(rb + v) * 3 + 2];
      rn[v] = r0 * r0 + r1 * r1 + r2 * r2;
    }
  }

  float acc = 0.0f;
  const int ct0 = chunk * kChunk;
  for (int t = 0; t < kChunk; ++t) {
    const int c  = (ct0 + t) * 16 + l15;
    const float b0 = B[c * 3 + 0];
    const float b1 = B[c * 3 + 1];
    const float b2 = B[c * 3 + 2];
    const float cn = b0 * b0 + b1 * b1 + b2 * b2;

    // B-matrix 4x16 f32: lanes 0-15 {K0,K1}, lanes 16-31 {K2,pad}, N = lane&15.
    v2f b;
    b.x = hi ? b2 : b0;
    b.y = hi ? 0.0f : b1;

    v8f cz = {};
    // D = A x B (+0): 16x16 tile of f32 dot products.
    v8f d = __builtin_amdgcn_wmma_f32_16x16x4_f32(
        /*neg_a=*/false, a, /*neg_b=*/false, b,
        /*c_mod=*/(short)0, cz, /*reuse_a=*/false, /*reuse_b=*/false);

#pragma unroll
    for (int v = 0; v < 8; ++v) {
      const float d2 = rn[v] + cn - 2.0f * d[v];
      acc += __builtin_amdgcn_exp2f(d2 * expScale);  // v_exp_f32 (TRANS pipe)
    }
  }

  // In-wave butterfly reduction (wave32), deterministic.
#pragma unroll
  for (int off = 16; off > 0; off >>= 1) acc += __shfl_xor(acc, off, 32);
  if (lane == 0) partials[wave] = acc;
}

// Single block: deterministic tree reduction of the 3x4096 partials + final math.
__global__ __launch_bounds__(256)
void itl_finalize(const float* __restrict__ partials,
                  const float* __restrict__ ks_p,
                  const float* __restrict__ theta_p,
                  float* __restrict__ out) {
  __shared__ float sm[3][256];
  const int tid = threadIdx.x;
#pragma unroll
  for (int m = 0; m < 3; ++m) {
    float s = 0.0f;
    for (int i = tid; i < kJobsPerMat; i += 256) s += partials[m * kJobsPerMat + i];
    sm[m][tid] = s;
  }
  __syncthreads();
  for (int st = 128; st > 0; st >>= 1) {
    if (tid < st) {
      sm[0][tid] += sm[0][tid + st];
      sm[1][tid] += sm[1][tid + st];
      sm[2][tid] += sm[2][tid + st];
    }
    __syncthreads();
  }
  if (tid == 0) {
    const float ks    = ks_p[0];
    const float theta = theta_p[0];
    // mean = sum / N^2 / sqrt(2*pi*ks)
    const float norm = (1.0f / ((float)kN * (float)kN)) *
                       rsqrtf(6.2831853071795864f * ks);
    const float mxx = sm[0][0] * norm;
    const float mzz = sm[1][0] * norm;
    const float mxz = sm[2][0] * norm;
    const float r   = logf(sqrtf(mxx * mzz + 1e-5f) / (mxz + 1e-5f));
    out[2 * kN * kD] = r * theta;
  }
}

// Pass-through outputs: d_out[0:24576]=x, d_out[24576:49152]=z.
__global__ __launch_bounds__(256)
void itl_copy(const float* __restrict__ X, const float* __restrict__ Z,
              float* __restrict__ out) {
  const int i = blockIdx.x * blockDim.x + threadIdx.x;
  const int n = kN * kD;
  if (i < n)
    out[i] = X[i];
  else if (i < 2 * n)
    out[i] = Z[i - n];
}

extern "C" void kernel_launch(void* const* d_in, const int* in_sizes, int n_in,
                              void* d_out, int out_size, void* d_ws, size_t ws_size,
                              hipStream_t stream) {
  (void)in_sizes; (void)n_in; (void)out_size; (void)ws_size;
  const float* x     = (const float*)d_in[0];
  const float* z     = (const float*)d_in[1];
  const float* ks    = (const float*)d_in[2];
  const float* theta = (const float*)d_in[3];
  float* out      = (float*)d_out;
  float* partials = (float*)d_ws;  // kTotalJobs floats = 48 KB

  itl_copy<<<(2 * kN * kD + 255) / 256, 256, 0, stream>>>(x, z, out);
  itl_gram_accum<<<kGrid, kBlock, 0, stream>>>(x, z, ks, partials);
  itl_finalize<<<1, 256, 0, stream>>>(partials, ks, theta, out);
}